// Model_58085137711859
// MI455X (gfx1250) — compile-verified
//
#include <hip/hip_runtime.h>

typedef __attribute__((ext_vector_type(2))) float v2f;
typedef __attribute__((ext_vector_type(8))) float v8f;

namespace {
constexpr int THREADS = 256;   // 8 wave32
constexpr int B_DIM   = 64;    // batch
constexpr int SEQ     = 720;
constexpr int F_DIM   = 321;   // features (grid dim)
constexpr int O_DIM   = 336;   // pred_len
constexpr int ICH     = 752;   // 720 + 32 time-embed rows

// LDS layout (float offsets). Pitches chosen so (pitch mod 64)/4 is odd ->
// conflict-free ds_load_b64 fragment reads across 16 lanes.
constexpr int APITCH1 = 756;               // 752 padded, 16B-aligned rows
constexpr int HPITCH  = 340;               // 336 padded
constexpr int AS_OFF  = 0;                 // A1: 64*756        = 48384 f
constexpr int H1_OFF  = B_DIM * APITCH1;   // h1: 64*340        = 21760 f
constexpr int H2_OFF  = 0;                 // h2 reuses dead A1 region
constexpr int WS_OFF  = H1_OFF + B_DIM * HPITCH;  // weight chunk: <= 6144 f
constexpr int SL_OFF  = WS_OFF + 6144;     // seq_last[64]
constexpr int LDS_FLOATS = SL_OFF + B_DIM; // 76352 floats = 305408 B < 320 KB
}

// One grouped-linear layer: [64 x I] (LDS) @ W[f]^T ([O_DIM x I] row-major,
// streamed from HBM through LDS) + bias -> LDS h or global out.
template <bool TO_GLOBAL>
__device__ __forceinline__ void grouped_layer(
    float* lds, int a_off, int apitch, int I, int KC,
    const float* __restrict__ Wg, const float* __restrict__ bg,
    int h_off, float* __restrict__ outg, int f)
{
  const int tid  = threadIdx.x;
  const int lane = tid & 31;
  const int wid  = tid >> 5;
  const int mo   = wid & 3;       // batch tile 0..3 (rows 16*mo..)
  const int ng   = wid >> 2;      // 0/1: which of the paired output tiles
  const int ln   = lane & 15;
  const int lh   = lane >> 4;     // lane half selects K pair (f32 16x16x4 layout)
  const int wpitch = (KC & 7) ? KC : KC + 4;  // avoid power-of-2 bank stride
  float* Ws = lds + WS_OFF;

  for (int pt = 0; pt < O_DIM / 32 + 1; ++pt) {   // 11 tile-pairs cover 21 tiles
    const int  o0    = (2 * pt + ng) * 16;
    const bool valid = (o0 < O_DIM);
    v8f acc = {};
    for (int k0 = 0; k0 < I; k0 += KC) {
      // Cooperative stage of 32 weight rows x KC cols (coalesced; read once).
      for (int j = tid; j < 32 * KC; j += THREADS) {
        int r = j / KC, c = j - r * KC;
        int orow = 32 * pt + r; if (orow > O_DIM - 1) orow = O_DIM - 1;
        Ws[r * wpitch + c] = Wg[orow * I + k0 + c];
        if (c == 0 && k0 + KC < I)                     // prefetch next chunk
          __builtin_prefetch(&Wg[orow * I + k0 + KC], 0, 1);
      }
      __syncthreads();
      const float* Arow = lds + a_off + (mo * 16 + ln) * apitch + k0 + 2 * lh;
      const float* Brow = Ws + (ng * 16 + ln) * wpitch + 2 * lh;
      for (int kk = 0; kk < KC; kk += 4) {
        v2f a = *(const v2f*)(Arow + kk);   // A[m, k..k+1] per f32 16x16x4 layout
        v2f b = *(const v2f*)(Brow + kk);   // B[k..k+1, n] (W^T)
        acc = __builtin_amdgcn_wmma_f32_16x16x4_f32(
            false, a, false, b, (short)0, acc, false, false);
      }
      __syncthreads();
    }
    if (valid) {
      const float bias = bg[o0 + ln];
      const int   o    = o0 + ln;
      #pragma unroll
      for (int v = 0; v < 8; ++v) {         // D layout: VGPR v -> M=v / v+8
        const int   m   = mo * 16 + v + lh * 8;
        const float val = acc[v] + bias;
        if (TO_GLOBAL)
          outg[(m * O_DIM + o) * F_DIM + f] = val + lds[SL_OFF + m];
        else
          lds[h_off + m * HPITCH + o] = val;
      }
    }
  }
}

__global__ void __launch_bounds__(THREADS)
tlinear_fused(const float* __restrict__ x, const long long* __restrict__ ts,
              const float* __restrict__ W1, const float* __restrict__ b1,
              const float* __restrict__ W2, const float* __restrict__ b2,
              const float* __restrict__ W3, const float* __restrict__ b3,
              const float* __restrict__ dayt, const float* __restrict__ hourt,
              float* __restrict__ out)
{
  extern __shared__ float lds[];
  const int f   = blockIdx.x;
  const int tid = threadIdx.x;

  // seq_last for this feature
  if (tid < B_DIM)
    lds[SL_OFF + tid] = x[((size_t)tid * SEQ + (SEQ - 1)) * F_DIM + f];
  __syncthreads();

  // Build A1 = [x - seq_last ; time_embed] (64 x 752) in LDS.
  for (int idx = tid; idx < B_DIM * ICH; idx += THREADS) {
    const int b = idx / ICH, i = idx - b * ICH;
    float v;
    if (i < SEQ) {
      v = x[((size_t)b * SEQ + i) * F_DIM + f] - lds[SL_OFF + b];
    } else {
      const int j = i - SEQ;
      const long long d = ts[((size_t)b * SEQ + (SEQ - 1)) * 2 + 0];
      const long long h = ts[((size_t)b * SEQ + (SEQ - 1)) * 2 + 1];
      v = (j < 16) ? dayt[d * 16 + j] : hourt[h * 16 + (j - 16)];
    }
    lds[AS_OFF + b * APITCH1 + i] = v;
  }
  __syncthreads();

  grouped_layer<false>(lds, AS_OFF, APITCH1, ICH, 188,
                       W1 + (size_t)f * O_DIM * ICH, b1 + (size_t)f * O_DIM,
                       H1_OFF, nullptr, f);
  __syncthreads();
  grouped_layer<false>(lds, H1_OFF, HPITCH, O_DIM, 168,
                       W2 + (size_t)f * O_DIM * O_DIM, b2 + (size_t)f * O_DIM,
                       H2_OFF, nullptr, f);
  __syncthreads();
  grouped_layer<true>(lds, H2_OFF, HPITCH, O_DIM, 168,
                      W3 + (size_t)f * O_DIM * O_DIM, b3 + (size_t)f * O_DIM,
                      -1, out, f);
}

extern "C" void kernel_launch(void* const* d_in, const int* in_sizes, int n_in,
                              void* d_out, int out_size, void* d_ws, size_t ws_size,
                              hipStream_t stream) {
  (void)in_sizes; (void)n_in; (void)out_size; (void)d_ws; (void)ws_size;
  const float*     x     = (const float*)d_in[0];
  const long long* ts    = (const long long*)d_in[1];
  const float*     W1    = (const float*)d_in[2];
  const float*     b1    = (const float*)d_in[3];
  const float*     W2    = (const float*)d_in[4];
  const float*     b2    = (const float*)d_in[5];
  const float*     W3    = (const float*)d_in[6];
  const float*     b3    = (const float*)d_in[7];
  const float*     dayt  = (const float*)d_in[8];
  const float*     hourt = (const float*)d_in[9];
  float*           out   = (float*)d_out;

  const size_t shmem = (size_t)LDS_FLOATS * sizeof(float);
  hipFuncSetAttribute((const void*)tlinear_fused,
                      hipFuncAttributeMaxDynamicSharedMemorySize, (int)shmem);
  tlinear_fused<<<F_DIM, THREADS, shmem, stream>>>(
      x, ts, W1, b1, W2, b2, W3, b3, dayt, hourt, out);
}